// DGCNN_TNet_90082644066705
// MI455X (gfx1250) — compile-verified
//
#include <hip/hip_runtime.h>
#include <hip/hip_bf16.h>

typedef __attribute__((ext_vector_type(2))) float v2f;
typedef __attribute__((ext_vector_type(8))) float v8f;

#define BATCH 8
#define NPTS  4096
#define KNN   20
#define PPB   8                 // points per edgeconv block
#define ROWS  (PPB * KNN)       // 160 (n,k)-rows per block -> 10 M-tiles of 16

// ---------------------------------------------------------------------------
// Stage 1: exact KNN (top-20 smallest squared distances, includes self).
// Whole batch point cloud staged in LDS (48KB); one thread per query point,
// register-resident sorted insertion (fully unrolled -> stays in VGPRs).
// ---------------------------------------------------------------------------
__global__ void __launch_bounds__(256) knn_kernel(const float* __restrict__ x,
                                                  int* __restrict__ idx) {
  __shared__ float xs[NPTS], ys[NPTS], zs[NPTS];
  const int b = blockIdx.x >> 4;       // 16 chunks of 256 queries per batch
  const int chunk = blockIdx.x & 15;
  const float* xb = x + b * 3 * NPTS;  // x is (B, 3, N)
  for (int i = threadIdx.x; i < NPTS; i += 256) {
    xs[i] = xb[i];
    ys[i] = xb[NPTS + i];
    zs[i] = xb[2 * NPTS + i];
  }
  __syncthreads();
  const int q = chunk * 256 + threadIdx.x;
  const float qx = xs[q], qy = ys[q], qz = zs[q];
  float best[KNN];
  int bidx[KNN];
#pragma unroll
  for (int j = 0; j < KNN; ++j) { best[j] = 3.4e38f; bidx[j] = 0; }
  for (int m = 0; m < NPTS; ++m) {
    const float dx = xs[m] - qx, dy = ys[m] - qy, dz = zs[m] - qz;
    const float d = dx * dx + dy * dy + dz * dz;
    if (d < best[KNN - 1]) {           // rare after warm-up
      float nd = d; int ni = m;
#pragma unroll
      for (int j = 0; j < KNN; ++j) {  // strict < keeps lower index on ties
        if (nd < best[j]) {
          const float td = best[j]; const int ti = bidx[j];
          best[j] = nd; bidx[j] = ni; nd = td; ni = ti;
        }
      }
    }
  }
  int* op = idx + ((size_t)b * NPTS + q) * KNN;
#pragma unroll
  for (int j = 0; j < KNN; ++j) op[j] = bidx[j];
}

// ---------------------------------------------------------------------------
// Stage 2: fused edge-feature + conv1(6->64) + conv2(64->128, WMMA) + max_k.
// Block = 8 points = 160 rows. h1 built scalar (K=6 is negligible), h2 done
// as 10x8 tiles of v_wmma_f32_16x16x4_f32 over K=64, max over k folded via
// LDS int-atomicMax on non-negative relu outputs.
// ---------------------------------------------------------------------------
__global__ void __launch_bounds__(256) edgeconv_kernel(
    const float* __restrict__ x, const int* __restrict__ idx,
    const float* __restrict__ W1, const float* __restrict__ b1,
    const float* __restrict__ W2, const float* __restrict__ b2,
    float* __restrict__ feat) {
  __shared__ float e[ROWS][6];         // edge features
  __shared__ float h1[ROWS * 64];      // conv1 output (40KB)
  __shared__ float W1s[64 * 6];
  __shared__ float b1s[64];
  __shared__ float b2s[128];
  __shared__ float fmax_[PPB * 128];   // running max over k

  const int tid = threadIdx.x;
  const int b = blockIdx.x >> 9;              // 512 blocks per batch
  const int pbase = (blockIdx.x & 511) * PPB;
  const float* xb = x + b * 3 * NPTS;

  for (int i = tid; i < 64 * 6; i += 256) W1s[i] = W1[i];
  if (tid < 64)  b1s[tid] = b1[tid];
  if (tid < 128) b2s[tid] = b2[tid];
  for (int i = tid; i < PPB * 128; i += 256) fmax_[i] = 0.0f;

  if (tid < ROWS) {
    const int p = tid / KNN, j = tid - p * KNN;
    const int n = pbase + p;
    const int nb = idx[((size_t)b * NPTS + n) * KNN + j];
#pragma unroll
    for (int c = 0; c < 3; ++c) {
      const float cen = xb[c * NPTS + n];
      const float nbr = xb[c * NPTS + nb];
      e[tid][c] = cen;                 // central
      e[tid][3 + c] = nbr - cen;       // neighbor - central
    }
  }
  __syncthreads();

  // h1 = relu(e @ W1^T + b1)
  for (int t = tid; t < ROWS * 64; t += 256) {
    const int r = t >> 6, o = t & 63;
    float acc = b1s[o];
#pragma unroll
    for (int c = 0; c < 6; ++c) acc += e[r][c] * W1s[o * 6 + c];
    h1[t] = fmaxf(acc, 0.0f);
  }
  __syncthreads();

  // h2 = relu(h1 @ W2^T + b2) with WMMA f32 16x16x4; fold max over k.
  const int wave = tid >> 5;
  const int lane = tid & 31;
  const int half = lane >> 4;          // K-pair select per ISA f32 A layout
  const int l = lane & 15;
  for (int tile = wave; tile < (ROWS / 16) * 8; tile += 8) {  // 10 per wave
    const int m0 = (tile >> 3) * 16;
    const int n0 = (tile & 7) * 16;
    v8f acc = {};
    const float* arow = &h1[(m0 + l) * 64];
    const float* brow = &W2[(n0 + l) * 64];   // B = W2^T, W2 is (128,64)
#pragma unroll
    for (int kk = 0; kk < 16; ++kk) {
      const int k0 = kk * 4 + 2 * half;
      v2f a, bb;
      a.x = arow[k0];  a.y = arow[k0 + 1];
      bb.x = brow[k0]; bb.y = brow[k0 + 1];
      acc = __builtin_amdgcn_wmma_f32_16x16x4_f32(false, a, false, bb,
                                                  (short)0, acc, false, false);
    }
    const float bias = b2s[n0 + l];
#pragma unroll
    for (int i = 0; i < 8; ++i) {      // lane holds rows m0+half*8+i, col n0+l
      const float v = fmaxf(acc[i] + bias, 0.0f);
      const int r = m0 + half * 8 + i;
      const int p = r / KNN;
      atomicMax((int*)&fmax_[p * 128 + (n0 + l)], __float_as_int(v));
    }
  }
  __syncthreads();

  for (int i = tid; i < PPB * 128; i += 256) {
    const int p = i >> 7, ch = i & 127;
    feat[(((size_t)b * NPTS) + pbase + p) * 128 + ch] = fmax_[p * 128 + ch];
  }
}

__global__ void zero_kernel(float* __restrict__ g, int n) {
  const int i = blockIdx.x * 256 + threadIdx.x;
  if (i < n) g[i] = 0.0f;
}

// ---------------------------------------------------------------------------
// Stage 3: conv3 (128->1024) via WMMA + global max over N points.
// One 16x16 C tile per wave, K=128 -> 32 wmma steps. Relu >= 0 so the
// cross-block max is an int atomicMax on float bits (g pre-zeroed).
// ---------------------------------------------------------------------------
__global__ void __launch_bounds__(256) conv3_kernel(
    const float* __restrict__ feat, const float* __restrict__ W3,
    const float* __restrict__ b3, float* __restrict__ g) {
  const int tid = threadIdx.x;
  const int wave = tid >> 5, lane = tid & 31;
  const int half = lane >> 4, l = lane & 15;
  const int b = blockIdx.x >> 11;            // 2048 tile-groups per batch
  const int rest = blockIdx.x & 2047;
  const int tile = rest * 8 + wave;          // 16384 tiles per batch
  const int m0 = (tile >> 6) * 16;           // 256 M-tiles
  const int n0 = (tile & 63) * 16;           // 64  N-tiles
  const float* arow = feat + (((size_t)b * NPTS) + m0 + l) * 128;
  const float* brow = W3 + (size_t)(n0 + l) * 128;  // B = W3^T, W3 is (1024,128)
  v8f acc = {};
#pragma unroll 8
  for (int kk = 0; kk < 32; ++kk) {
    const int k0 = kk * 4 + 2 * half;
    v2f a, bb;
    a.x = arow[k0];  a.y = arow[k0 + 1];
    bb.x = brow[k0]; bb.y = brow[k0 + 1];
    acc = __builtin_amdgcn_wmma_f32_16x16x4_f32(false, a, false, bb,
                                                (short)0, acc, false, false);
  }
  const float bias = b3[n0 + l];
  float lmax = 0.0f;
#pragma unroll
  for (int i = 0; i < 8; ++i) lmax = fmaxf(lmax, fmaxf(acc[i] + bias, 0.0f));
  atomicMax((int*)&g[b * 1024 + n0 + l], __float_as_int(lmax));
}

// ---------------------------------------------------------------------------
// Stage 4: tiny FC head (8x: 1024->512->256->9) + identity, single block.
// ---------------------------------------------------------------------------
__global__ void __launch_bounds__(256) fc_kernel(
    const float* __restrict__ g,
    const float* __restrict__ W4, const float* __restrict__ b4,
    const float* __restrict__ W5, const float* __restrict__ b5,
    const float* __restrict__ W6, const float* __restrict__ b6,
    float* __restrict__ out) {
  __shared__ float gs[1024], h4[512], h5[256];
  const int tid = threadIdx.x;
  for (int b = 0; b < BATCH; ++b) {
    for (int i = tid; i < 1024; i += 256) gs[i] = g[b * 1024 + i];
    __syncthreads();
    for (int o = tid; o < 512; o += 256) {
      float acc = b4[o];
      const float* w = W4 + (size_t)o * 1024;
      for (int c = 0; c < 1024; ++c) acc += gs[c] * w[c];
      h4[o] = fmaxf(acc, 0.0f);
    }
    __syncthreads();
    if (tid < 256) {
      float acc = b5[tid];
      const float* w = W5 + (size_t)tid * 512;
      for (int c = 0; c < 512; ++c) acc += h4[c] * w[c];
      h5[tid] = fmaxf(acc, 0.0f);
    }
    __syncthreads();
    if (tid < 9) {
      float acc = b6[tid];
      const float* w = W6 + tid * 256;
      for (int c = 0; c < 256; ++c) acc += h5[c] * w[c];
      const int i = tid / 3, j = tid - 3 * (tid / 3);
      out[b * 9 + tid] = acc + ((i == j) ? 1.0f : 0.0f);
    }
    __syncthreads();
  }
}

extern "C" void kernel_launch(void* const* d_in, const int* in_sizes, int n_in,
                              void* d_out, int out_size, void* d_ws, size_t ws_size,
                              hipStream_t stream) {
  const float* x  = (const float*)d_in[0];
  const float* W1 = (const float*)d_in[1];
  const float* b1 = (const float*)d_in[2];
  const float* W2 = (const float*)d_in[3];
  const float* b2 = (const float*)d_in[4];
  const float* W3 = (const float*)d_in[5];
  const float* b3 = (const float*)d_in[6];
  const float* W4 = (const float*)d_in[7];
  const float* b4 = (const float*)d_in[8];
  const float* W5 = (const float*)d_in[9];
  const float* b5 = (const float*)d_in[10];
  const float* W6 = (const float*)d_in[11];
  const float* b6 = (const float*)d_in[12];
  float* out = (float*)d_out;

  // Workspace layout: idx (2.62MB) | feat (16.78MB) | g (32KB)
  char* ws = (char*)d_ws;
  int* idx = (int*)ws;
  float* feat = (float*)(ws + (size_t)BATCH * NPTS * KNN * sizeof(int));
  float* g = feat + (size_t)BATCH * NPTS * 128;

  knn_kernel<<<BATCH * 16, 256, 0, stream>>>(x, idx);
  edgeconv_kernel<<<BATCH * (NPTS / PPB), 256, 0, stream>>>(x, idx, W1, b1, W2, b2, feat);
  zero_kernel<<<(BATCH * 1024 + 255) / 256, 256, 0, stream>>>(g, BATCH * 1024);
  conv3_kernel<<<BATCH * 2048, 256, 0, stream>>>(feat, W3, b3, g);
  fc_kernel<<<1, 256, 0, stream>>>(g, W4, b4, W5, b5, W6, b6, out);
}